// PEPS_45045617000638
// MI455X (gfx1250) — compile-verified
//
#include <hip/hip_runtime.h>
#include <math.h>

typedef __attribute__((ext_vector_type(2))) float v2f;
typedef __attribute__((ext_vector_type(8))) float v8f;

#define NROWS 6
#define NCOLS 6
#define DBOND 6
#define BATCH 8
#define MAXSTATE 279936   // 6^7 complex elements: max intermediate state size
#define PADC    1024      // per-buffer complex padding so A-tile overreads stay in-bounds
#define BUFC    (MAXSTATE + PADC)
#define NTHREADS 512

struct TensPtrs { const double* p[NROWS*NCOLS]; };

__host__ __device__ constexpr int cpow(int b, int e) { return e == 0 ? 1 : b * cpow(b, e - 1); }

struct Ctx {
  const int* x;
  const TensPtrs* tens;
  int s, tid, wave, nwaves, lane;
};

// One site absorption: S'[Mtot x 2N] = S[Mtot x 2K] * WR[2K x 2N] with all dims constexpr.
template<int r, int c>
__device__ __forceinline__ void site_step(const Ctx& ctx, float (*WR)[80],
                                          float2*& bin, float2*& bout)
{
  constexpr int U = (r == 0) ? 1 : DBOND;          // up-leg dim
  constexpr int D = (r == NROWS - 1) ? 1 : DBOND;  // down-leg dim
  constexpr int L = (c == 0) ? 1 : DBOND;
  constexpr int R = (c == NCOLS - 1) ? 1 : DBOND;
  constexpr int K = L * U;                         // complex contraction dim (l,u)
  constexpr int N = D * R;                         // complex output dim (d,r)
  constexpr int twoN = 2 * N;
  constexpr int P = cpow(D, c);                    // d-legs already emitted
  constexpr int Rest2 = (c < NCOLS - 1) ? cpow(U, NCOLS - 2 - c) : 1;
  constexpr int Unext = (c < NCOLS - 1) ? U : 1;
  constexpr int Pp = P * D;
  constexpr int Kout = R * Unext;                  // next state's complex K
  constexpr int Mtot = cpow(U, NCOLS - 1 - c) * P; // state rows
  constexpr int K4C = (2 * K + 3) / 4;
  constexpr int Mtiles = (Mtot + 15) / 16;
  constexpr int Ntiles = (twoN + 15) / 16;
  constexpr bool MFULL = (Mtot % 16) == 0;
  constexpr int WRROWS = 4 * K4C;                  // LDS rows actually read

  const int tid = ctx.tid;

  // ---- build real-expanded site matrix in LDS (zero-padded WRROWS x 80) ----
  for (int i = tid; i < WRROWS * 80; i += NTHREADS) (&WR[0][0])[i] = 0.f;
  __syncthreads();
  {
    const double* tp = ctx.tens->p[r * NCOLS + c];
    const int psel   = ctx.x[(ctx.s * NROWS + r) * NCOLS + c];
    for (int i = tid; i < K * N; i += NTHREADS) {
      const int k = i / N, n = i - k * N;
      const int l = k / U, u = k - l * U;
      const int dd = n / R, rr = n - dd * R;
      const int ci = (((psel * U + u) * D + dd) * L + l) * R + rr;
      const float re = (float)tp[2 * ci];
      const float im = (float)tp[2 * ci + 1];
      WR[k][n]         = re;
      WR[k][n + N]     = im;
      WR[k + K][n]     = -im;
      WR[k + K][n + N] = re;
    }
  }
  __syncthreads();

  const float* fin = (const float*)bin;
  float* fout      = (float*)bout;
  const int lane  = ctx.lane;
  const int lan15 = lane & 15;
  const int kb    = (lane >= 16) ? 2 : 0;

  for (int mt = ctx.wave; mt < Mtiles; mt += ctx.nwaves) {
    const int mrow = (mt << 4) + lan15;
    const int base = 2 * mrow * K;

    v2f afrag[K4C];
#pragma unroll
    for (int k4 = 0; k4 < K4C; ++k4) {
      const int j0 = (k4 << 2) + kb;
      const int j1 = j0 + 1;
      afrag[k4].x = fin[base + ((j0 < K) ? (j0 << 1) : (((j0 - K) << 1) + 1))];
      afrag[k4].y = fin[base + ((j1 < K) ? (j1 << 1) : (((j1 - K) << 1) + 1))];
    }

    const int mbase = (mt << 4) + ((lane >= 16) ? 8 : 0);

#pragma unroll
    for (int nt = 0; nt < Ntiles; ++nt) {
      const int ncol = (nt << 4) + lan15;
      const bool nok = ((nt + 1) * 16 <= twoN) || (ncol < twoN);  // folds for full tiles

      // Phase 1: gather all B fragments (clustered ds_loads, single wait)
      v2f bfrag[K4C];
#pragma unroll
      for (int k4 = 0; k4 < K4C; ++k4) {
        const int j0 = (k4 << 2) + kb;
        bfrag[k4].x = WR[j0][ncol];
        bfrag[k4].y = WR[j0 + 1][ncol];
      }

      // Phase 2: back-to-back WMMA accumulation chain
      v8f acc = {};
#pragma unroll
      for (int k4 = 0; k4 < K4C; ++k4) {
        acc = __builtin_amdgcn_wmma_f32_16x16x4_f32(
            false, afrag[k4], false, bfrag[k4], (short)0, acc, false, false);
      }

      // Phase 3: scatter into next-state layout (constexpr divisors -> shifts/magic)
      const int isIm  = (ncol >= N) ? 1 : 0;
      const int n_c   = isIm ? (ncol - N) : ncol;
      const int dd    = n_c / R, rr = n_c - dd * R;
      const int k2b   = rr * Unext;
#pragma unroll
      for (int v = 0; v < 8; ++v) {
        const int mr = mbase + v;
        bool ok = nok;
        if constexpr (!MFULL) ok = ok && (mr < Mtot);
        if (ok) {
          const int rest = mr / P,       pidx  = mr - rest * P;
          const int un   = rest / Rest2, rest2 = rest - un * Rest2;
          const int m2   = rest2 * Pp + pidx * D + dd;
          const int k2   = k2b + un;
          fout[((m2 * Kout + k2) << 1) + isIm] = acc[v];
        }
      }
    }
  }
  __threadfence();
  __syncthreads();
  { float2* t = bin; bin = bout; bout = t; }
}

// Between rows: max-abs rescale (log accumulated in f64) + boundary relayout.
__device__ __forceinline__ void row_transition(const Ctx& ctx, float* red,
                                               float* scalp, double* lscp,
                                               float2*& bin, float2*& bout)
{
  constexpr int Mtot  = cpow(DBOND, NCOLS);  // 46656 boundary entries, K==1
  constexpr int Mhigh = Mtot / DBOND;        // 7776
  const int tid = ctx.tid;
  const float* fb = (const float*)bin;
  float mx = 0.f;
  for (int i = tid; i < Mtot * 2; i += NTHREADS) mx = fmaxf(mx, fabsf(fb[i]));
  red[tid] = mx;
  __syncthreads();
  for (int st = NTHREADS / 2; st > 0; st >>= 1) {
    if (tid < st) red[tid] = fmaxf(red[tid], red[tid + st]);
    __syncthreads();
  }
  if (tid == 0) {
    float m = red[0];
    if (!(m > 0.f)) m = 1.f;
    *scalp = 1.f / m;
    *lscp += log((double)m);
  }
  __syncthreads();
  const float sc = *scalp;
  for (int i = tid; i < Mtot; i += NTHREADS) {
    const int u0 = i / Mhigh;          // constexpr divisor
    const int m  = i - u0 * Mhigh;
    float2 vv = bin[i];
    vv.x *= sc; vv.y *= sc;
    bout[m * DBOND + u0] = vv;         // next state S0[m][k=u0], K=6
  }
  __threadfence();
  __syncthreads();
  { float2* t = bin; bin = bout; bout = t; }
}

template<int r, int c> struct SiteSeq {
  static __device__ __forceinline__ void run(const Ctx& ctx, float (*WR)[80],
                                             float2*& bin, float2*& bout) {
    site_step<r, c>(ctx, WR, bin, bout);
    SiteSeq<r, c + 1>::run(ctx, WR, bin, bout);
  }
};
template<int r> struct SiteSeq<r, NCOLS> {
  static __device__ __forceinline__ void run(const Ctx&, float (*)[80], float2*&, float2*&) {}
};

template<int r> struct RowSeq {
  static __device__ __forceinline__ void run(const Ctx& ctx, float (*WR)[80], float* red,
                                             float* scalp, double* lscp,
                                             float2*& bin, float2*& bout) {
    SiteSeq<r, 0>::run(ctx, WR, bin, bout);
    if constexpr (r < NROWS - 1) {
      row_transition(ctx, red, scalp, lscp, bin, bout);
    }
    RowSeq<r + 1>::run(ctx, WR, red, scalp, lscp, bin, bout);
  }
};
template<> struct RowSeq<NROWS> {
  static __device__ __forceinline__ void run(const Ctx&, float (*)[80], float*, float*,
                                             double*, float2*&, float2*&) {}
};

__global__ void __launch_bounds__(NTHREADS)
peps_logamp_kernel(const int* __restrict__ x, TensPtrs tens,
                   float* __restrict__ out, int out_size,
                   float2* __restrict__ ws)
{
  __shared__ float  WR[72][80];
  __shared__ float  red[NTHREADS];
  __shared__ float  scal;
  __shared__ double lsc;

  Ctx ctx;
  ctx.x = x;
  ctx.tens = &tens;
  ctx.s = blockIdx.x;
  ctx.tid = threadIdx.x;
  ctx.lane = ctx.tid & 31;
  ctx.wave = ctx.tid >> 5;
  ctx.nwaves = NTHREADS / 32;

  float2* bin  = ws + (size_t)ctx.s * (2 * BUFC);
  float2* bout = bin + BUFC;

  if (ctx.tid == 0) { lsc = 0.0; bin[0] = make_float2(1.f, 0.f); }
  __syncthreads();

  RowSeq<0>::run(ctx, WR, red, &scal, &lsc, bin, bout);

  if (ctx.tid == 0) {
    const double ar = (double)bin[0].x;
    const double ai = (double)bin[0].y;
    const double lr = 0.5 * log(ar * ar + ai * ai) + lsc;
    const double li = atan2(ai, ar);
    if (out_size >= 2 * BATCH) {
      out[2 * ctx.s]     = (float)lr;
      out[2 * ctx.s + 1] = (float)li;
    } else if (ctx.s < out_size) {
      out[ctx.s] = (float)lr;
    }
  }
}

extern "C" void kernel_launch(void* const* d_in, const int* in_sizes, int n_in,
                              void* d_out, int out_size, void* d_ws, size_t ws_size,
                              hipStream_t stream) {
  (void)in_sizes; (void)ws_size;
  const int* x = (const int*)d_in[0];
  TensPtrs tp;
  for (int i = 0; i < NROWS * NCOLS; ++i) {
    const int idx = 1 + i;
    tp.p[i] = (idx < n_in) ? (const double*)d_in[idx]
                           : (const double*)d_in[n_in - 1];
  }
  peps_logamp_kernel<<<dim3(BATCH), dim3(NTHREADS), 0, stream>>>(
      x, tp, (float*)d_out, out_size, (float2*)d_ws);
}